// MambaLayer3D_19241453486104
// MI455X (gfx1250) — compile-verified
//
#include <hip/hip_runtime.h>
#include <math.h>

#define L 4096
#define CDIM 256
#define DIN 512
#define DSTATE 16
#define NDIR 3

typedef float v2f __attribute__((ext_vector_type(2)));
typedef float v8f __attribute__((ext_vector_type(8)));

__device__ __forceinline__ float silu_f(float x) { return x / (1.0f + expf(-x)); }
__device__ __forceinline__ float softplus_f(float x) {
  return (x > 20.0f) ? x : log1pf(expf(x));
}

// voxel index v (in x's (d,h,w) flattening) for sequence position l of direction dir
__device__ __forceinline__ int voxel_of(int dir, int l) {
  if (dir == 0) return l;                      // l = dd*256 + h*16 + w
  int a = l >> 8, b = (l >> 4) & 15, c = l & 15;
  if (dir == 1) return a * 256 + c * 16 + b;   // l = dd*256 + w*16 + h
  return c * 256 + a * 16 + b;                 // l = h*256 + w*16 + dd
}

// ---------------- LayerNorm + gather into per-direction sequences ----------------
__global__ void ln_gather_kernel(const float* __restrict__ x,
                                 const float* __restrict__ lnw,
                                 const float* __restrict__ lnb,
                                 float* __restrict__ seq) {
  int gl = blockIdx.x;           // 0 .. 3*4096-1
  int dir = gl >> 12;
  int l = gl & (L - 1);
  int c = threadIdx.x;           // 0..255 (channel)
  int v = voxel_of(dir, l);
  float xv = x[c * L + v];
  __shared__ float s_sum[256], s_sq[256];
  s_sum[c] = xv; s_sq[c] = xv * xv;
  __syncthreads();
  for (int off = 128; off > 0; off >>= 1) {
    if (c < off) { s_sum[c] += s_sum[c + off]; s_sq[c] += s_sq[c + off]; }
    __syncthreads();
  }
  float mu  = s_sum[0] * (1.0f / 256.0f);
  float var = s_sq[0] * (1.0f / 256.0f) - mu * mu;
  float inv = rsqrtf(var + 1e-5f);
  seq[(size_t)gl * CDIM + c] = (xv - mu) * inv * lnw[c] + lnb[c];
}

// ---------------- Generic fp32 WMMA GEMM: C[M,N] = A[M,K] @ W[N,K]^T ----------------
// One wave computes a 16 x (NT*16) tile; NT is compile-time so every WMMA is
// unconditional. Fragment fetch is software-pipelined (double-buffered) so the
// global_load_b64s for k+4 are in flight while the WMMAs for k execute.
// Block = 128 thr (4 waves, each its own M row-tile). grid.y = M/64, grid.x = N/(NT*16).
// MODE 1: C = softplus(C + bias[col])
template <int NT, int MODE>
__global__ __launch_bounds__(128) void wmma_gemm_kernel(
    const float* __restrict__ A, int lda,
    const float* __restrict__ W, int ldw,
    float* __restrict__ C, int ldc,
    int K, const float* __restrict__ bias) {
  int wave = threadIdx.x >> 5;
  int lane = threadIdx.x & 31;
  int half = lane >> 4;          // 0: K even pair / M low 8 ; 1: K odd pair / M high 8
  int r = lane & 15;
  int m0 = (blockIdx.y * 4 + wave) * 16;
  int nb = blockIdx.x * (NT * 16);

  const float* arow = A + (size_t)(m0 + r) * lda + 2 * half;
  const float* wbase = W + (size_t)(nb + r) * ldw + 2 * half;

  v8f acc[NT];
#pragma unroll
  for (int t = 0; t < NT; ++t)
#pragma unroll
    for (int i = 0; i < 8; ++i) acc[t][i] = 0.0f;

  // prologue: fragments for k0 = 0
  v2f a_cur = *(const v2f*)(arow);
  v2f b_cur[NT];
#pragma unroll
  for (int t = 0; t < NT; ++t)
    b_cur[t] = *(const v2f*)(wbase + (size_t)(t * 16) * ldw);

#pragma unroll 2
  for (int k0 = 0; k0 < K - 4; k0 += 4) {
    // issue next k-step's loads before consuming the current fragments
    v2f a_nxt = *(const v2f*)(arow + k0 + 4);
    v2f b_nxt[NT];
#pragma unroll
    for (int t = 0; t < NT; ++t)
      b_nxt[t] = *(const v2f*)(wbase + (size_t)(t * 16) * ldw + k0 + 4);

#pragma unroll
    for (int t = 0; t < NT; ++t)
      acc[t] = __builtin_amdgcn_wmma_f32_16x16x4_f32(
          false, a_cur, false, b_cur[t], (short)0, acc[t], false, false);

    a_cur = a_nxt;
#pragma unroll
    for (int t = 0; t < NT; ++t) b_cur[t] = b_nxt[t];
  }

  // epilogue k-step
#pragma unroll
  for (int t = 0; t < NT; ++t)
    acc[t] = __builtin_amdgcn_wmma_f32_16x16x4_f32(
        false, a_cur, false, b_cur[t], (short)0, acc[t], false, false);

#pragma unroll
  for (int t = 0; t < NT; ++t) {
    int col = nb + t * 16 + r;
#pragma unroll
    for (int i = 0; i < 8; ++i) {
      int row = m0 + i + half * 8;   // C layout: VGPR i, lanes<16 M=i, lanes>=16 M=i+8
      float vv = acc[t][i];
      if (MODE == 1) vv = softplus_f(vv + bias[col]);
      C[(size_t)row * ldc + col] = vv;
    }
  }
}

// ---------------- Causal depthwise conv1d (k=4) + SiLU, fwd + reversed branch ----------------
// xs layout: [branch][dir][t][d]  (branch-major so each branch's rows are contiguous for GEMM)
__global__ void conv_silu_kernel(const float* __restrict__ xz,
                                 const float* __restrict__ cw_f, const float* __restrict__ cb_f,
                                 const float* __restrict__ cw_b, const float* __restrict__ cb_b,
                                 float* __restrict__ xs) {
  long tid = (long)blockIdx.x * blockDim.x + threadIdx.x;   // 0 .. 6*4096*512-1
  int d = (int)(tid & (DIN - 1));
  int t = (int)((tid >> 9) & (L - 1));
  int q = (int)(tid >> 21);           // 0..5 = br*3 + dir
  int br = q / 3, dir = q % 3;
  const float* cw = br ? cw_b : cw_f;
  const float* cb = br ? cb_b : cb_f;
  float acc = cb[d];
#pragma unroll
  for (int k = 0; k < 4; ++k) {
    int j = t - 3 + k;                // causal pad-left 3
    if (j >= 0) {
      int srcl = br ? (L - 1 - j) : j;                     // bwd branch scans reversed seq
      acc += cw[d * 4 + k] * xz[((size_t)(dir * L + srcl)) * 1024 + d];  // xb = cols 0..511
    }
  }
  xs[tid] = silu_f(acc);
}

__global__ void zero_kernel(float* __restrict__ p, long n) {
  long i = (long)blockIdx.x * blockDim.x + threadIdx.x;
  if (i < n) p[i] = 0.0f;
}

// ---------------- Selective scan: 6 scans (3 dir x 2 branch), 1 lane per (channel,state) ----
// block = 256 thr (8 waves x 2 channels x 16 states); grid = (32 ch-groups, 2 branch, 3 dir)
__global__ void scan_kernel(const float* __restrict__ dtp,   // [br][dir][t][512]
                            const float* __restrict__ dbl,   // [br][dir][t][48]
                            const float* __restrict__ xs,    // [br][dir][t][512]
                            const float* __restrict__ Alog_f, const float* __restrict__ Alog_b,
                            const float* __restrict__ Dsk_f,  const float* __restrict__ Dsk_b,
                            float* __restrict__ ycomb) {     // [dir][l][512]
  int br = blockIdx.y, dir = blockIdx.z;
  int q = br * 3 + dir;
  int lane = threadIdx.x & 31;
  int wave = threadIdx.x >> 5;
  int s = lane & 15;
  int ch = blockIdx.x * 16 + wave * 2 + (lane >> 4);
  const float* Alog = br ? Alog_b : Alog_f;
  const float* Dsk  = br ? Dsk_b  : Dsk_f;
  float Aval = -expf(Alog[ch * DSTATE + s]);
  float Dv = Dsk[ch];
  const float* dtp_p = dtp + (size_t)q * L * DIN;
  const float* dbl_p = dbl + (size_t)q * L * 48;
  const float* xs_p  = xs  + (size_t)q * L * DIN;
  float* y_p = ycomb + (size_t)dir * L * DIN;

  float h = 0.0f;
  for (int t = 0; t < L; ++t) {
    float dtv = dtp_p[(size_t)t * DIN + ch];
    float xsv = xs_p[(size_t)t * DIN + ch];
    float Bv  = dbl_p[t * 48 + 16 + s];
    float Cv  = dbl_p[t * 48 + 32 + s];
    float dA  = expf(dtv * Aval);
    h = dA * h + dtv * Bv * xsv;
    float p = h * Cv;
    p += __shfl_xor(p, 1, 32);
    p += __shfl_xor(p, 2, 32);
    p += __shfl_xor(p, 4, 32);
    p += __shfl_xor(p, 8, 32);
    if (s == 0) {
      int outl = br ? (L - 1 - t) : t;   // bwd branch un-reversed on write
      atomicAdd(&y_p[(size_t)outl * DIN + ch], p + Dv * xsv);
    }
  }
}

// ---------------- y = (y_f + y_b) * silu(z), in place over ycomb ----------------
__global__ void combine_kernel(float* __restrict__ ycomb, const float* __restrict__ xz) {
  long i = (long)blockIdx.x * blockDim.x + threadIdx.x;  // [dir][l][512]
  int d = (int)(i & (DIN - 1));
  long row = i >> 9;                                     // dir*L + l
  float z = xz[row * 1024 + 512 + d];
  ycomb[i] = ycomb[i] * silu_f(z);
}

// ---------------- out = x + scatter of the 3 directions' out_proj results ----------------
__global__ void final_kernel(const float* __restrict__ x, const float* __restrict__ contrib,
                             float* __restrict__ out) {
  int i = blockIdx.x * blockDim.x + threadIdx.x;  // c*4096 + v
  int v = i & (L - 1);
  int c = i >> 12;
  int dd = v >> 8, hh = (v >> 4) & 15, ww = v & 15;
  int l0 = v;
  int l1 = dd * 256 + ww * 16 + hh;
  int l2 = hh * 256 + ww * 16 + dd;
  out[i] = x[i]
         + contrib[((size_t)(0 * L + l0)) * CDIM + c]
         + contrib[((size_t)(1 * L + l1)) * CDIM + c]
         + contrib[((size_t)(2 * L + l2)) * CDIM + c];
}

extern "C" void kernel_launch(void* const* d_in, const int* in_sizes, int n_in,
                              void* d_out, int out_size, void* d_ws, size_t ws_size,
                              hipStream_t stream) {
  (void)in_sizes; (void)n_in; (void)out_size; (void)ws_size;
  const float* x          = (const float*)d_in[0];
  const float* ln_w       = (const float*)d_in[1];
  const float* ln_b       = (const float*)d_in[2];
  const float* in_proj_w  = (const float*)d_in[3];
  const float* out_proj_w = (const float*)d_in[4];
  const float* conv_w     = (const float*)d_in[5];
  const float* conv_b     = (const float*)d_in[6];
  const float* x_proj_w   = (const float*)d_in[7];
  const float* dt_proj_w  = (const float*)d_in[8];
  const float* dt_proj_b  = (const float*)d_in[9];
  const float* A_log      = (const float*)d_in[10];
  const float* D_skip     = (const float*)d_in[11];
  const float* conv_w_b   = (const float*)d_in[12];
  const float* conv_b_b   = (const float*)d_in[13];
  const float* x_proj_w_b = (const float*)d_in[14];
  const float* dt_proj_w_b= (const float*)d_in[15];
  const float* dt_proj_b_b= (const float*)d_in[16];
  const float* A_log_b    = (const float*)d_in[17];
  const float* D_skip_b   = (const float*)d_in[18];
  float* out = (float*)d_out;
  float* ws  = (float*)d_ws;

  const long M3 = 3L * L;   // 12288 rows (3 directions batched, shared weights)

  // workspace layout (floats); `seq` is reused as the out_proj result buffer
  float* seq   = ws;                           // M3*256
  float* xz    = seq  + (size_t)M3 * CDIM;     // M3*1024
  float* xs    = xz   + (size_t)M3 * 1024;     // 2*M3*512  [branch][dir][t][d]
  float* dblb  = xs   + (size_t)2 * M3 * DIN;  // 2*M3*48
  float* dtp   = dblb + (size_t)2 * M3 * 48;   // 2*M3*512
  float* ycomb = dtp  + (size_t)2 * M3 * DIN;  // M3*512

  // 1. gather + layernorm (3 directions)
  ln_gather_kernel<<<dim3((unsigned)M3), dim3(256), 0, stream>>>(x, ln_w, ln_b, seq);

  // 2. in_proj: (12288 x 256) @ (1024 x 256)^T -> xz   (N=1024 -> 16 blocks of 64)
  wmma_gemm_kernel<4, 0><<<dim3(1024 / 64, (unsigned)(M3 / 64)), dim3(128), 0, stream>>>(
      seq, CDIM, in_proj_w, CDIM, xz, 1024, CDIM, nullptr);

  // 3. causal conv + SiLU for both branches
  conv_silu_kernel<<<dim3((unsigned)(2 * M3 * DIN / 256)), dim3(256), 0, stream>>>(
      xz, conv_w, conv_b, conv_w_b, conv_b_b, xs);

  // 4. x_proj: (12288 x 512) @ (48 x 512)^T per branch   (N=48 -> NT=3, exact)
  wmma_gemm_kernel<3, 0><<<dim3(1, (unsigned)(M3 / 64)), dim3(128), 0, stream>>>(
      xs, DIN, x_proj_w, DIN, dblb, 48, DIN, nullptr);
  wmma_gemm_kernel<3, 0><<<dim3(1, (unsigned)(M3 / 64)), dim3(128), 0, stream>>>(
      xs + (size_t)M3 * DIN, DIN, x_proj_w_b, DIN, dblb + (size_t)M3 * 48, 48, DIN,
      nullptr);

  // 5. dt_proj (+bias, softplus): (12288 x 16) @ (512 x 16)^T per branch
  wmma_gemm_kernel<4, 1><<<dim3(512 / 64, (unsigned)(M3 / 64)), dim3(128), 0, stream>>>(
      dblb, 48, dt_proj_w, 16, dtp, DIN, 16, dt_proj_b);
  wmma_gemm_kernel<4, 1><<<dim3(512 / 64, (unsigned)(M3 / 64)), dim3(128), 0, stream>>>(
      dblb + (size_t)M3 * 48, 48, dt_proj_w_b, 16, dtp + (size_t)M3 * DIN, DIN, 16,
      dt_proj_b_b);

  // 6. zero the branch-accumulation buffer (graph replays: must re-init every call)
  zero_kernel<<<dim3((unsigned)(M3 * DIN / 256)), dim3(256), 0, stream>>>(ycomb, M3 * DIN);

  // 7. selective scans: 6 scans fully parallel, atomic-accumulate fwd+bwd (+D skip)
  scan_kernel<<<dim3(32, 2, 3), dim3(256), 0, stream>>>(
      dtp, dblb, xs, A_log, A_log_b, D_skip, D_skip_b, ycomb);

  // 8. gate with silu(z), in place
  combine_kernel<<<dim3((unsigned)(M3 * DIN / 256)), dim3(256), 0, stream>>>(ycomb, xz);

  // 9. out_proj: (12288 x 512) @ (256 x 512)^T -> contrib (reuse seq buffer)
  wmma_gemm_kernel<4, 0><<<dim3(256 / 64, (unsigned)(M3 / 64)), dim3(128), 0, stream>>>(
      ycomb, DIN, out_proj_w, DIN, seq, CDIM, DIN, nullptr);

  // 10. residual + 3-direction scatter-sum
  final_kernel<<<dim3(CDIM * L / 256), dim3(256), 0, stream>>>(x, seq, out);
}